// Simplicial_01_sparse_Layer_90065464197253
// MI455X (gfx1250) — compile-verified
//
#include <hip/hip_runtime.h>
#include <math.h>

typedef __attribute__((ext_vector_type(2))) float v2f;
typedef __attribute__((ext_vector_type(8))) float v8f;

#define DM 128      // model dim
#define NHEAD 8
// dh = 16, 1/sqrt(dh) = 0.25

// ---------------------------------------------------------------- fill zero
__global__ void fill0_kernel(float* __restrict__ p, long long n) {
  long long i = (long long)blockIdx.x * blockDim.x + threadIdx.x;
  long long st = (long long)gridDim.x * blockDim.x;
  for (; i < n; i += st) p[i] = 0.f;
}

// ---------------------------------------------------------------- WMMA GEMM
// C[N x M] = A[N x K] @ B[K x M] (+bias) (+relu), all row-major fp32.
// 256 threads = 8 waves. blockIdx.x -> 16-row tile; (blockIdx.y*8 + wave) -> 16-col tile.
// A tile staged in LDS (padded), B streamed from global (L2-resident weights).
template<int K, int M, bool BIAS, bool RELU>
__global__ void wmma_gemm_kernel(const float* __restrict__ A,
                                 const float* __restrict__ B,
                                 const float* __restrict__ bias,
                                 float* __restrict__ C) {
  constexpr int KPAD = K + 4;
  __shared__ float sA[16 * KPAD];
  const int rowTile = blockIdx.x * 16;

  for (int idx = threadIdx.x; idx < 16 * K; idx += 256) {
    int r = idx / K, c = idx - r * K;
    sA[r * KPAD + c] = A[(long long)(rowTile + r) * K + c];
  }
  __syncthreads();

  const int wave = threadIdx.x >> 5;
  const int lane = threadIdx.x & 31;
  const int colBase = (blockIdx.y * 8 + wave) * 16;
  const int rA = lane & 15;            // A-fragment row
  const int kb = (lane >> 4) << 1;     // 0 for lanes 0-15, 2 for lanes 16-31
  const int cB = colBase + (lane & 15);

  v8f acc = {};
  #pragma unroll
  for (int k = 0; k < K; k += 4) {
    v2f a, b;
    a.x = sA[rA * KPAD + k + kb];
    a.y = sA[rA * KPAD + k + kb + 1];
    b.x = B[(long long)(k + kb) * M + cB];
    b.y = B[(long long)(k + kb + 1) * M + cB];
    acc = __builtin_amdgcn_wmma_f32_16x16x4_f32(
        false, a, false, b, (short)0, acc, false, false);
  }

  const int rowOff = rowTile + ((lane >> 4) << 3);  // +8 for upper lane half
  float bv = BIAS ? bias[cB] : 0.f;
  #pragma unroll
  for (int r = 0; r < 8; ++r) {
    float v = acc[r] + bv;
    if (RELU) v = fmaxf(v, 0.f);
    C[(long long)(rowOff + r) * M + cB] = v;
  }
}

// ---------------------------------------------------------------- edge attention
// One wave32 per edge. Lane l owns elements [4l, 4l+4) of the 128-wide row.
// Head h = l/4 ; per-head reduction across 4 lanes via shfl_xor.
__global__ void edge_attn_kernel(const long long* __restrict__ ei,
                                 const float* __restrict__ Q,
                                 const float* __restrict__ Km,
                                 const float* __restrict__ V,
                                 const float* __restrict__ Eh,
                                 float* __restrict__ wV,
                                 float* __restrict__ Z,
                                 int E) {
  const int lane = threadIdx.x & 31;
  long long w = (long long)blockIdx.x * (blockDim.x >> 5) + (threadIdx.x >> 5);
  long long stride = (long long)gridDim.x * (blockDim.x >> 5);
  for (long long e = w; e < E; e += stride) {
    const long long src = ei[e];
    const long long dst = ei[E + e];
    const float4 k4 = ((const float4*)(Km + src * DM))[lane];
    const float4 q4 = ((const float4*)(Q + dst * DM))[lane];
    const float4 e4 = ((const float4*)(Eh + e * DM))[lane];
    float part = (k4.x * q4.x * e4.x + k4.y * q4.y * e4.y +
                  k4.z * q4.z * e4.z + k4.w * q4.w * e4.w) * 0.25f;
    part += __shfl_xor(part, 1, 32);
    part += __shfl_xor(part, 2, 32);          // now per-head sum in all 4 lanes
    float s = __expf(fminf(fmaxf(part, -5.f), 5.f));
    const float4 v4 = ((const float4*)(V + src * DM))[lane];
    float* out = wV + dst * DM + lane * 4;
    atomicAdd(out + 0, v4.x * s);
    atomicAdd(out + 1, v4.y * s);
    atomicAdd(out + 2, v4.z * s);
    atomicAdd(out + 3, v4.w * s);
    if ((lane & 3) == 0) atomicAdd(Z + dst * NHEAD + (lane >> 2), s);
  }
}

// ---------------------------------------------------------------- residual + column stats
__global__ void attn_resid_stats_kernel(const float* __restrict__ x,
                                        const float* __restrict__ wV,
                                        const float* __restrict__ Z,
                                        float* __restrict__ y,
                                        float* __restrict__ stats,  // [128] sum | [128] sumsq
                                        long long total) {
  __shared__ float ssum[DM], ssq[DM];
  if (threadIdx.x < DM) { ssum[threadIdx.x] = 0.f; ssq[threadIdx.x] = 0.f; }
  __syncthreads();
  long long i = (long long)blockIdx.x * blockDim.x + threadIdx.x;
  long long st = (long long)gridDim.x * blockDim.x;
  for (; i < total; i += st) {
    int col = (int)(i & (DM - 1));
    long long row = i >> 7;
    float z = Z[row * NHEAD + (col >> 4)];
    float v = x[i] + wV[i] / (z + 1e-6f);
    y[i] = v;
    atomicAdd(&ssum[col], v);
    atomicAdd(&ssq[col], v * v);
  }
  __syncthreads();
  if (threadIdx.x < DM) {
    atomicAdd(&stats[threadIdx.x], ssum[threadIdx.x]);
    atomicAdd(&stats[DM + threadIdx.x], ssq[threadIdx.x]);
  }
}

__global__ void add_resid_stats_kernel(const float* __restrict__ a,
                                       const float* __restrict__ b,
                                       float* __restrict__ y,
                                       float* __restrict__ stats,
                                       long long total) {
  __shared__ float ssum[DM], ssq[DM];
  if (threadIdx.x < DM) { ssum[threadIdx.x] = 0.f; ssq[threadIdx.x] = 0.f; }
  __syncthreads();
  long long i = (long long)blockIdx.x * blockDim.x + threadIdx.x;
  long long st = (long long)gridDim.x * blockDim.x;
  for (; i < total; i += st) {
    int col = (int)(i & (DM - 1));
    float v = a[i] + b[i];
    y[i] = v;
    atomicAdd(&ssum[col], v);
    atomicAdd(&ssq[col], v * v);
  }
  __syncthreads();
  if (threadIdx.x < DM) {
    atomicAdd(&stats[threadIdx.x], ssum[threadIdx.x]);
    atomicAdd(&stats[DM + threadIdx.x], ssq[threadIdx.x]);
  }
}

// ---------------------------------------------------------------- BN coeff + apply
__global__ void bn_coeff_kernel(const float* __restrict__ stats,
                                const float* __restrict__ gamma,
                                const float* __restrict__ beta,
                                float* __restrict__ coef, float invN) {
  int c = threadIdx.x;
  if (c < DM) {
    float mu = stats[c] * invN;
    float var = stats[DM + c] * invN - mu * mu;
    float a = gamma[c] * rsqrtf(var + 1e-5f);
    coef[c] = a;
    coef[DM + c] = beta[c] - a * mu;
  }
}

__global__ void bn_apply_kernel(const float* __restrict__ y,
                                const float* __restrict__ coef,
                                float* __restrict__ out, long long total) {
  long long i = (long long)blockIdx.x * blockDim.x + threadIdx.x;
  long long st = (long long)gridDim.x * blockDim.x;
  for (; i < total; i += st) {
    int col = (int)(i & (DM - 1));
    out[i] = coef[col] * y[i] + coef[DM + col];
  }
}

// ---------------------------------------------------------------- launch
extern "C" void kernel_launch(void* const* d_in, const int* in_sizes, int n_in,
                              void* d_out, int out_size, void* d_ws, size_t ws_size,
                              hipStream_t stream) {
  const float*     x     = (const float*)d_in[0];
  const long long* eidx  = (const long long*)d_in[1];
  const float*     eattr = (const float*)d_in[2];
  const float*     WQ    = (const float*)d_in[3];
  const float*     WK    = (const float*)d_in[4];
  const float*     WV    = (const float*)d_in[5];
  const float*     WE    = (const float*)d_in[6];
  const float*     g1    = (const float*)d_in[7];
  const float*     be1   = (const float*)d_in[8];
  const float*     g2    = (const float*)d_in[9];
  const float*     be2   = (const float*)d_in[10];
  const float*     W1    = (const float*)d_in[11];
  const float*     b1    = (const float*)d_in[12];
  const float*     W2    = (const float*)d_in[13];
  const float*     b2    = (const float*)d_in[14];
  float* out = (float*)d_out;

  const int N = in_sizes[0] / DM;   // 50000
  const int E = in_sizes[2] / DM;   // 600000
  const long long ND = (long long)N * DM;

  // workspace layout (floats)
  float* ws = (float*)d_ws;
  float* Q      = ws;                    // ND
  float* Kb     = Q + ND;                // ND
  float* Vb     = Kb + ND;               // ND
  float* wV     = Vb + ND;               // ND
  float* Zb     = wV + ND;               // N*8
  float* stats1 = Zb + (long long)N * NHEAD;  // 256
  float* coef1  = stats1 + 256;          // 256
  float* stats2 = coef1 + 256;           // 256
  float* coef2  = stats2 + 256;          // 256
  float* Eh     = coef2 + 256;           // E*128  (reused as ff1: N*256)
  // reuse:
  float* y1  = Q;    // after edge_attn, Q dead
  float* hbn = Kb;   // after edge_attn, K dead
  float* ff1 = Eh;   // after edge_attn, Eh dead
  float* ff2 = Vb;   // after FF1 input read, V dead
  float* y2  = wV;   // after attn finalize, wV dead

  // 1) zero accumulators: wV .. coef2 (contiguous)
  {
    long long zn = ND + (long long)N * NHEAD + 1024;
    fill0_kernel<<<1024, 256, 0, stream>>>(wV, zn);
  }

  // 2) dense projections (fp32 WMMA)
  dim3 blk(256);
  wmma_gemm_kernel<DM, DM, false, false><<<dim3(N / 16, 1), blk, 0, stream>>>(x, WQ, nullptr, Q);
  wmma_gemm_kernel<DM, DM, false, false><<<dim3(N / 16, 1), blk, 0, stream>>>(x, WK, nullptr, Kb);
  wmma_gemm_kernel<DM, DM, false, false><<<dim3(N / 16, 1), blk, 0, stream>>>(x, WV, nullptr, Vb);
  wmma_gemm_kernel<DM, DM, false, false><<<dim3(E / 16, 1), blk, 0, stream>>>(eattr, WE, nullptr, Eh);

  // 3) edge-sparse attention (wave per edge)
  {
    int blocks = (E + 7) / 8;
    edge_attn_kernel<<<blocks, 256, 0, stream>>>(eidx, Q, Kb, Vb, Eh, wV, Zb, E);
  }

  // 4) h_attn finalize + residual + BN1 stats
  attn_resid_stats_kernel<<<1024, 256, 0, stream>>>(x, wV, Zb, y1, stats1, ND);
  bn_coeff_kernel<<<1, 128, 0, stream>>>(stats1, g1, be1, coef1, 1.0f / (float)N);
  bn_apply_kernel<<<1024, 256, 0, stream>>>(y1, coef1, hbn, ND);

  // 5) FF block (WMMA): ff1 = relu(hbn@W1 + b1) ; ff2 = ff1@W2 + b2
  wmma_gemm_kernel<DM, 2 * DM, true, true><<<dim3(N / 16, 2), blk, 0, stream>>>(hbn, W1, b1, ff1);
  wmma_gemm_kernel<2 * DM, DM, true, false><<<dim3(N / 16, 1), blk, 0, stream>>>(ff1, W2, b2, ff2);

  // 6) residual + BN2 -> output
  add_resid_stats_kernel<<<1024, 256, 0, stream>>>(hbn, ff2, y2, stats2, ND);
  bn_coeff_kernel<<<1, 128, 0, stream>>>(stats2, g2, be2, coef2, 1.0f / (float)N);
  bn_apply_kernel<<<1024, 256, 0, stream>>>(y2, coef2, out, ND);
}